// OntologyHierarchicalAttention_82231443849242
// MI455X (gfx1250) — compile-verified
//
#include <hip/hip_runtime.h>
#include <hip/hip_bf16.h>
#include <math.h>

// OntologyHierarchicalAttention for MI455X (gfx1250), wave32 + WMMA fp32 path
// + async global->LDS tile staging (ASYNCcnt) with double buffering.
//
// B=4, N=1024, C=1024, H=16, d=64, NUM_RELATIONS=100
// All GEMMs use V_WMMA_F32_16X16X4_F32 (fp32 in/out -> matches reference dtype;
// logits are multiplied by relation ids up to 99 before softmax, so low-precision
// A/B inputs would get amplified -- fp32 matrix path is the right choice here).
//
// Workspace layout (floats):
//   [0..127]          rel_means (100 used)
//   [256 ..)          q  [B,H,N,64]  (pre-scaled by 1/sqrt(d))
//   +4M               k  [B,H,N,64]
//   +4M               v  [B,H,N,64]
//   +4M               ao [B,N,C]     (heads merged back, proj input)
// total ~64 MB.

typedef __attribute__((ext_vector_type(2))) float v2f;
typedef __attribute__((ext_vector_type(8))) float v8f;
// Explicit LDS-address-space float: guarantees ds_load/ds_store codegen (the
// dynamic extern __shared__ base otherwise loses addrspace inference and the
// hot-loop fragment reads lowered to flat_load_b64).
typedef __attribute__((address_space(3))) float lds_f;

#define NEG_BIG (-1e9f)
#define QK_SCALE 0.125f // 64^-0.5

static __device__ __forceinline__ v8f wmma_f32(v2f a, v2f b, v8f c) {
  // D = A(16x4) * B(4x16) + C(16x16), fp32.
  // 8 args: (neg_a, A, neg_b, B, c_mod, C, reuse_a, reuse_b)
  return __builtin_amdgcn_wmma_f32_16x16x4_f32(false, a, false, b, (short)0, c,
                                               false, false);
}

// --- CDNA5 async global->LDS copy (per-lane, 16B), tracked by ASYNCcnt -----
// An AS3 pointer IS the wave-relative 32-bit LDS byte address.
static __device__ __forceinline__ void async_ld_b128(lds_f* lds_dst,
                                                     const float* gsrc) {
#if defined(__HIP_DEVICE_COMPILE__)
  unsigned lo = (unsigned)(unsigned long long)lds_dst;
  asm volatile("global_load_async_to_lds_b128 %0, %1, off"
               :
               : "v"(lo), "v"(gsrc)
               : "memory");
#endif
}
// generic-pointer overload (static __shared__ in the GEMM kernels); the LDS
// aperture keeps the byte offset in the low 32 bits of the flat address.
static __device__ __forceinline__ void async_ld_b128(void* lds_dst,
                                                     const void* gsrc) {
#if defined(__HIP_DEVICE_COMPILE__)
  unsigned lo = (unsigned)(unsigned long long)lds_dst;
  asm volatile("global_load_async_to_lds_b128 %0, %1, off"
               :
               : "v"(lo), "v"(gsrc)
               : "memory");
#endif
}
static __device__ __forceinline__ void wait_async0() {
#if defined(__HIP_DEVICE_COMPILE__)
  asm volatile("s_wait_asynccnt 0x0" ::: "memory");
#endif
}

// ---------------------------------------------------------------------------
// Kernel 0: rel_means[r] = mean(rel_emb[r, 0:64])
// ---------------------------------------------------------------------------
__global__ void relmeans_kernel(const float* __restrict__ rel_emb,
                                float* __restrict__ relm) {
  int t = threadIdx.x;
  if (t < 100) {
    float s = 0.f;
    #pragma unroll
    for (int j = 0; j < 64; ++j) s += rel_emb[t * 64 + j];
    relm[t] = s * (1.0f / 64.0f);
  }
}

// ---------------------------------------------------------------------------
// Kernels 1 & 3: OUT[M, Nout] = A[M,1024] @ W[Nout,1024]^T + bias
// Block: 256 threads = 8 waves, tile 64(M) x 128(N). Wave (mw, nw) owns a
// 16x64 strip -> 4 fp32 WMMA accumulators. K staged through double-buffered
// LDS in chunks of 16 via async global->LDS b128 copies (stride 20 -> both
// 16B-aligned chunks and conflict-free fragment gathers).
// EPI==0: scatter into q/k/v head layout [B,H,N,64], q pre-scaled.
// EPI==1: plain store to out0[M,1024].
// ---------------------------------------------------------------------------
template <int EPI>
__global__ __launch_bounds__(256) void gemm_wmma_kernel(
    const float* __restrict__ A, const float* __restrict__ W,
    const float* __restrict__ bias, float* __restrict__ out0,
    float* __restrict__ out1, float* __restrict__ out2) {
  __shared__ float As[2][64 * 20];   // 64 rows x 16 k, stride 20
  __shared__ float Ws[2][128 * 20];  // 128 rows x 16 k, stride 20

  const int t = threadIdx.x;
  const int lane = t & 31;
  const int w = t >> 5;
  const int mw = w & 3;       // wave M index: rows 16*mw..+15 of the tile
  const int nw = w >> 2;      // wave N index: cols 64*nw..+63 of the tile
  const int lo = lane & 15;   // fragment row/col within a 16-wide tile
  const int hi = lane >> 4;   // half-wave selector (K offset +2 for A/B frags)
  const int m0 = blockIdx.y * 64;
  const int n0 = blockIdx.x * 128;

  // per-thread async-copy coordinates (4-float = 16B chunks)
  const int ar = t >> 2, ak = (t & 3) * 4;  // A tile: 256 chunks
  const int wr0 = t >> 2, wk0 = (t & 3) * 4;
  const int wr1 = (t + 256) >> 2, wk1 = ((t + 256) & 3) * 4;  // W tile: 512

  v8f acc[4] = {};

  // prologue: stream in tile 0
  async_ld_b128(&As[0][ar * 20 + ak], &A[(size_t)(m0 + ar) * 1024 + 0 + ak]);
  async_ld_b128(&Ws[0][wr0 * 20 + wk0], &W[(size_t)(n0 + wr0) * 1024 + wk0]);
  async_ld_b128(&Ws[0][wr1 * 20 + wk1], &W[(size_t)(n0 + wr1) * 1024 + wk1]);

  for (int it = 0; it < 64; ++it) {
    const int cur = it & 1;
    wait_async0();
    __syncthreads();  // tile `cur` resident for all waves

    if (it < 63) {  // stream in next tile while computing this one
      int k0 = (it + 1) * 16;
      async_ld_b128(&As[cur ^ 1][ar * 20 + ak],
                    &A[(size_t)(m0 + ar) * 1024 + k0 + ak]);
      async_ld_b128(&Ws[cur ^ 1][wr0 * 20 + wk0],
                    &W[(size_t)(n0 + wr0) * 1024 + k0 + wk0]);
      async_ld_b128(&Ws[cur ^ 1][wr1 * 20 + wk1],
                    &W[(size_t)(n0 + wr1) * 1024 + k0 + wk1]);
    }

    const float* ap = &As[cur][(16 * mw + lo) * 20];
    #pragma unroll
    for (int kk = 0; kk < 16; kk += 4) {
      int koff = kk + 2 * hi;
      v2f a;
      a[0] = ap[koff];
      a[1] = ap[koff + 1];
      #pragma unroll
      for (int ns = 0; ns < 4; ++ns) {
        const float* bp = &Ws[cur][(64 * nw + 16 * ns + lo) * 20];
        v2f b;
        b[0] = bp[koff];
        b[1] = bp[koff + 1];
        acc[ns] = wmma_f32(a, b, acc[ns]);
      }
    }
    __syncthreads();  // all waves done reading tile `cur` before its re-fill
  }

  // Epilogue. C/D layout: VGPR e -> row e (lanes 0-15) / e+8 (lanes 16-31),
  // lane&15 -> column.
  #pragma unroll
  for (int ns = 0; ns < 4; ++ns) {
    int n = n0 + 64 * nw + 16 * ns + lo;
    float bv = bias[n];
    #pragma unroll
    for (int e = 0; e < 8; ++e) {
      int m = m0 + 16 * mw + e + 8 * hi;
      float val = acc[ns][e] + bv;
      if (EPI == 0) {
        int which = n >> 10;        // 0=q 1=k 2=v
        int c = n & 1023;
        int h = c >> 6, dd = c & 63;
        int b = m >> 10, nn = m & 1023;
        size_t idx = ((size_t)((b * 16 + h) * 1024 + nn)) * 64 + dd;
        if (which == 0)
          out0[idx] = val * QK_SCALE;
        else if (which == 1)
          out1[idx] = val;
        else
          out2[idx] = val;
      } else {
        out0[(size_t)m * 1024 + n] = val;
      }
    }
  }
}

// ---------------------------------------------------------------------------
// Kernel 2: attention for one (b,h) pair and one 16-row q tile.
// grid = (N/16, B*H), 256 threads = 8 waves.
// Dynamic LDS (~137 KB; CDNA5 WGP has 320 KB), all accesses through explicit
// AS3 pointers so everything lowers to ds ops:
//   S   : 16 x 1028  logits/probs panel     (stride 1028 -> conflict-free)
//   Qs  : 16 x 68    q tile                 (stride 68, 16B-aligned rows)
//   KT  : 2 x 128 x 68  double-buffered K tiles (buffer 0 reused as partials)
//   RM  : 128        rel_means
// K tiles are streamed with async global->LDS copies (ASYNCcnt) so tile jt+1
// transfers while WMMAs consume tile jt.
// ---------------------------------------------------------------------------
__global__ __launch_bounds__(256) void attn_kernel(
    const float* __restrict__ q, const float* __restrict__ k,
    const float* __restrict__ v, const int* __restrict__ relmask,
    const float* __restrict__ relm_g, float* __restrict__ ao) {
  extern __shared__ float smem_raw[];
  lds_f* smem = (lds_f*)smem_raw;
  lds_f* S = smem;                 // 16*1028
  lds_f* Qs = S + 16 * 1028;       // 16*68
  lds_f* KT0 = Qs + 16 * 68;       // 128*68
  lds_f* KT1 = KT0 + 128 * 68;     // 128*68
  lds_f* RM = KT1 + 128 * 68;      // 128
  lds_f* KTb[2] = {KT0, KT1};

  const int t = threadIdx.x;
  const int lane = t & 31;
  const int w = t >> 5;
  const int lo = lane & 15;
  const int hi = lane >> 4;
  const int i0 = blockIdx.x * 16;
  const int bh = blockIdx.y;
  const int b = bh >> 4, h = bh & 15;

  const float* qh = q + (size_t)bh * (1024 * 64);
  const float* kh = k + (size_t)bh * (1024 * 64);
  const float* vh = v + (size_t)bh * (1024 * 64);

  // async-stage q tile (16x64 = 256 x 16B chunks, one per thread)
  {
    int r = t >> 4, d4 = (t & 15) * 4;
    async_ld_b128(&Qs[r * 68 + d4], &qh[(size_t)(i0 + r) * 64 + d4]);
  }
  // async-stage K tile 0 (128x64 = 2048 chunks, 8 per thread)
  #pragma unroll
  for (int i = 0; i < 8; ++i) {
    int idx = t + 256 * i;
    int r = idx >> 4, d4 = (idx & 15) * 4;
    async_ld_b128(&KT0[r * 68 + d4], &kh[(size_t)r * 64 + d4]);
  }
  if (t < 100) RM[t] = relm_g[t];
  wait_async0();
  __syncthreads();

  // ---- Pass 1: S[i,j] = logits, WMMA q.k^T in 128-wide j tiles ----
  for (int jt = 0; jt < 8; ++jt) {
    const int cur = jt & 1;
    if (jt < 7) {  // stream next K tile while computing this one
      int jn = (jt + 1) * 128;
      lds_f* KTn = KTb[cur ^ 1];
      #pragma unroll
      for (int i = 0; i < 8; ++i) {
        int idx = t + 256 * i;
        int r = idx >> 4, d4 = (idx & 15) * 4;
        async_ld_b128(&KTn[r * 68 + d4], &kh[(size_t)(jn + r) * 64 + d4]);
      }
    }

    int j0 = jt * 128 + w * 16;     // this wave's 16 columns
    v8f acc = {};
    const lds_f* ap = &Qs[lo * 68];
    const lds_f* bp = &KTb[cur][(w * 16 + lo) * 68];
    #pragma unroll
    for (int kk = 0; kk < 64; kk += 4) {
      int koff = kk + 2 * hi;
      v2f a, bb;
      a[0] = ap[koff];
      a[1] = ap[koff + 1];
      bb[0] = bp[koff];
      bb[1] = bp[koff + 1];
      acc = wmma_f32(a, bb, acc);   // dots[i, j] (q pre-scaled)
    }

    int j = j0 + lo;
    #pragma unroll
    for (int e = 0; e < 8; ++e) {
      int il = e + 8 * hi;          // local row
      int gi = i0 + il;
      int mk = relmask[((size_t)(b * 1024 + gi)) * 1024 + j];
      float mv = (float)mk;         // hierarchy level, NOT 0/1
      float dot = acc[e];
      // dots + dots*bias, then *m + (1-m)*(-1e9)
      S[il * 1028 + j] = dot * (1.0f + RM[mk]) * mv + (1.0f - mv) * NEG_BIG;
    }
    wait_async0();    // next tile resident (this wave's copies)
    __syncthreads();  // ...and everyone's; current tile fully consumed
  }

  // ---- Pass 2: row softmax (wave handles 2 rows, 32-lane reductions) ----
  #pragma unroll
  for (int rr = 0; rr < 2; ++rr) {
    int row = 2 * w + rr;
    lds_f* sp = &S[row * 1028];
    float mx = -3.4e38f;
    for (int c2 = lane; c2 < 1024; c2 += 32) mx = fmaxf(mx, sp[c2]);
    #pragma unroll
    for (int o = 16; o > 0; o >>= 1) mx = fmaxf(mx, __shfl_xor(mx, o, 32));
    float sum = 0.f;
    for (int c2 = lane; c2 < 1024; c2 += 32) {
      float ev = __expf(sp[c2] - mx);
      sp[c2] = ev;
      sum += ev;
    }
    #pragma unroll
    for (int o = 16; o > 0; o >>= 1) sum += __shfl_xor(sum, o, 32);
    float inv = 1.0f / sum;
    for (int c2 = lane; c2 < 1024; c2 += 32) sp[c2] *= inv;
  }
  __syncthreads();

  // ---- Pass 3: out(16x64) = P(16x1024) @ V(1024x64), K split over waves ----
  v8f oacc[4] = {};
  {
    int kb = w * 128;               // this wave's K range
    const lds_f* ap = &S[lo * 1028];
    for (int kk = 0; kk < 128; kk += 4) {
      int kg = kb + kk + 2 * hi;
      v2f a;
      a[0] = ap[kg];
      a[1] = ap[kg + 1];
      #pragma unroll
      for (int ns = 0; ns < 4; ++ns) {
        v2f bb;  // V[j, dd]: lanes 0-15 -> consecutive dd (coalesced)
        bb[0] = vh[(size_t)kg * 64 + ns * 16 + lo];
        bb[1] = vh[(size_t)(kg + 1) * 64 + ns * 16 + lo];
        oacc[ns] = wmma_f32(a, bb, oacc[ns]);
      }
    }
  }

  // store partials into reused KT0 region, then sum the 8 wave partials
  lds_f* part = KT0;
  #pragma unroll
  for (int ns = 0; ns < 4; ++ns) {
    #pragma unroll
    for (int e = 0; e < 8; ++e) {
      int m = e + 8 * hi;
      part[w * 1024 + m * 64 + ns * 16 + lo] = oacc[ns][e];
    }
  }
  __syncthreads();

  #pragma unroll
  for (int i = 0; i < 4; ++i) {
    int e = t + 256 * i;            // 0..1023 -> (m, dd)
    int m = e >> 6, dd = e & 63;
    float s = 0.f;
    #pragma unroll
    for (int ww = 0; ww < 8; ++ww) s += part[ww * 1024 + e];
    // merged-head layout [B, N, H*64] == proj input
    ao[((size_t)(b * 1024 + i0 + m)) * 1024 + h * 64 + dd] = s;
  }
}

// ---------------------------------------------------------------------------
extern "C" void kernel_launch(void* const* d_in, const int* in_sizes, int n_in,
                              void* d_out, int out_size, void* d_ws,
                              size_t ws_size, hipStream_t stream) {
  const float* x = (const float*)d_in[0];
  const int* relations_mask = (const int*)d_in[1];
  const float* qkv_w = (const float*)d_in[2];
  const float* qkv_b = (const float*)d_in[3];
  const float* proj_w = (const float*)d_in[4];
  const float* proj_b = (const float*)d_in[5];
  const float* rel_emb = (const float*)d_in[6];
  float* out = (float*)d_out;

  float* ws = (float*)d_ws;
  const size_t HSZ = (size_t)4 * 16 * 1024 * 64;  // 4,194,304 floats
  float* relm = ws;
  float* q = ws + 256;
  float* k = q + HSZ;
  float* v = k + HSZ;
  float* ao = v + HSZ;

  // 0) relation means
  relmeans_kernel<<<1, 128, 0, stream>>>(rel_emb, relm);

  // 1) qkv projection: [4096,1024] x [1024,3072]^T, scatter to head layout
  gemm_wmma_kernel<0><<<dim3(24, 64), 256, 0, stream>>>(x, qkv_w, qkv_b, q, k,
                                                        v);

  // 2) attention (bias/mask/softmax fused), one block per (b,h, 16-row tile)
  size_t smem_bytes = (size_t)(16 * 1028 + 16 * 68 + 2 * 128 * 68 + 128) *
                      sizeof(float);  // ~137 KB, fits CDNA5 320 KB LDS
  attn_kernel<<<dim3(64, 64), 256, smem_bytes, stream>>>(q, k, v,
                                                         relations_mask, relm,
                                                         ao);

  // 3) output projection -> d_out
  gemm_wmma_kernel<1><<<dim3(8, 64), 256, 0, stream>>>(ao, proj_w, proj_b, out,
                                                       nullptr, nullptr);
}